// SENNLoss_75617194214078
// MI455X (gfx1250) — compile-verified
//
#include <hip/hip_runtime.h>

typedef float v2f __attribute__((ext_vector_type(2)));
typedef float v8f __attribute__((ext_vector_type(8)));

#define B_DIM 256
#define D_DIM 2048
#define K_DIM 64
#define NSPLIT 8
#define KCHUNK (D_DIM / NSPLIT)   // 256

#define WMMA_F32(a, b, c) \
    __builtin_amdgcn_wmma_f32_16x16x4_f32(false, (a), false, (b), (short)0, (c), false, false)

// ---------------------------------------------------------------------------
// Stage 1a: partial GEMM strip  Hp[split][m0:m0+16, 0:64] = X[:, kr] @ Wh[kr, :]
// One wave per 16x64 strip per K-split: 4 accumulators share each A-fragment.
// A-frag (16x4 f32): lane l -> row M = l%16, VGPR v -> K = v + 2*(l/16)
// ---------------------------------------------------------------------------
__global__ __launch_bounds__(32) void senn_h_partial(const float* __restrict__ X,
                                                     const float* __restrict__ Wh,
                                                     float* __restrict__ Hp) {
    const int lane = threadIdx.x;          // 0..31
    const int mr   = lane & 15;
    const int half = lane >> 4;            // 0 or 1
    const int m0   = blockIdx.x * 16;      // 16 row tiles (B=256)
    const int kbeg = blockIdx.y * KCHUNK;  // 8 K-splits

    // Incremented pointers -> constant IOFFSETs in the loop body.
    const float* xptr = X  + (size_t)(m0 + mr) * D_DIM + kbeg + 2 * half;
    const float* wptr = Wh + (size_t)(kbeg + 2 * half) * K_DIM + mr;

    v8f c0 = {}, c1 = {}, c2 = {}, c3 = {};
#pragma unroll 4
    for (int k = 0; k < KCHUNK; k += 4) {
        v2f a = *(const v2f*)xptr;                       // X[m, kk..kk+1]
        v2f b0, b1, b2, b3;                              // Wh[kk.., n-tile t]
        b0.x = wptr[0];  b0.y = wptr[K_DIM];
        b1.x = wptr[16]; b1.y = wptr[K_DIM + 16];
        b2.x = wptr[32]; b2.y = wptr[K_DIM + 32];
        b3.x = wptr[48]; b3.y = wptr[K_DIM + 48];
        c0 = WMMA_F32(a, b0, c0);
        c1 = WMMA_F32(a, b1, c1);
        c2 = WMMA_F32(a, b2, c2);
        c3 = WMMA_F32(a, b3, c3);
        xptr += 4;
        wptr += 4 * K_DIM;
    }

    // C/D layout: VGPR r, lane l -> M = r + 8*(l/16), N = l%16 (+16*t)
    float* dst = Hp + (size_t)blockIdx.y * (B_DIM * K_DIM) + (size_t)m0 * K_DIM;
#pragma unroll
    for (int r = 0; r < 8; ++r) {
        const int row = (r + 8 * half) * K_DIM + mr;
        dst[row +  0] = c0[r];
        dst[row + 16] = c1[r];
        dst[row + 32] = c2[r];
        dst[row + 48] = c3[r];
    }
}

// ---------------------------------------------------------------------------
// Stage 1b: combine K-splits, add bias, tanh -> H [256 x 64]
// ---------------------------------------------------------------------------
__global__ __launch_bounds__(256) void senn_h_combine(const float* __restrict__ Hp,
                                                      const float* __restrict__ bh,
                                                      float* __restrict__ H) {
    const int idx = blockIdx.x * blockDim.x + threadIdx.x;   // 0..16383
    if (idx >= B_DIM * K_DIM) return;
    float s = bh[idx & (K_DIM - 1)];
#pragma unroll
    for (int sp = 0; sp < NSPLIT; ++sp)
        s += Hp[(size_t)sp * (B_DIM * K_DIM) + idx];
    H[idx] = tanhf(s);
}

// ---------------------------------------------------------------------------
// Stage 2: G = H @ Wt^T (256 x 2048); per-wave 16x64 strip (4 accumulators),
// accumulate sum(G^2).  B[k, n] = Wt[n, k] -> contiguous float2 from Wt row n.
// ---------------------------------------------------------------------------
__global__ __launch_bounds__(32) void senn_g_kernel(const float* __restrict__ H,
                                                    const float* __restrict__ Wt,
                                                    float* __restrict__ partial) {
    const int lane = threadIdx.x;
    const int mr   = lane & 15;
    const int half = lane >> 4;
    const int m0   = blockIdx.x * 16;      // 16 strips over B
    const int n0   = blockIdx.y * 64;      // 32 strips over D

    const float* hptr = H  + (size_t)(m0 + mr) * K_DIM + 2 * half;
    const float* wptr = Wt + (size_t)(n0 + mr) * K_DIM + 2 * half;

    v8f c0 = {}, c1 = {}, c2 = {}, c3 = {};
#pragma unroll
    for (int k = 0; k < K_DIM; k += 4) {
        v2f a  = *(const v2f*)hptr;                      // H[m, kk..kk+1]
        v2f b0 = *(const v2f*)(wptr);                    // Wt row n0+mr
        v2f b1 = *(const v2f*)(wptr + 16 * K_DIM);       // Wt row n0+16+mr
        v2f b2 = *(const v2f*)(wptr + 32 * K_DIM);
        v2f b3 = *(const v2f*)(wptr + 48 * K_DIM);
        c0 = WMMA_F32(a, b0, c0);
        c1 = WMMA_F32(a, b1, c1);
        c2 = WMMA_F32(a, b2, c2);
        c3 = WMMA_F32(a, b3, c3);
        hptr += 4;
        wptr += 4;
    }

    float s = 0.0f;
#pragma unroll
    for (int r = 0; r < 8; ++r)
        s += c0[r] * c0[r] + c1[r] * c1[r] + c2[r] * c2[r] + c3[r] * c3[r];

    // wave32 reduction
#pragma unroll
    for (int off = 16; off > 0; off >>= 1)
        s += __shfl_down(s, off, 32);

    if (lane == 0)
        partial[blockIdx.y * gridDim.x + blockIdx.x] = s;
}

// ---------------------------------------------------------------------------
// Stage 3: deterministic reduction of strip partials -> scalar loss
// ---------------------------------------------------------------------------
__global__ __launch_bounds__(256) void senn_reduce(const float* __restrict__ partial,
                                                   float* __restrict__ out, int n) {
    __shared__ float sm[256];
    float s = 0.0f;
    for (int i = threadIdx.x; i < n; i += 256) s += partial[i];
    sm[threadIdx.x] = s;
    __syncthreads();
    for (int w = 128; w > 0; w >>= 1) {
        if (threadIdx.x < w) sm[threadIdx.x] += sm[threadIdx.x + w];
        __syncthreads();
    }
    if (threadIdx.x == 0)
        out[0] = sm[0] * (1.0f / (float)(B_DIM * D_DIM));
}

// ---------------------------------------------------------------------------
// Entry point. Inputs: x[256*2048], Wh[2048*64], bh[64], Wt[2048*64], bt[64]
// (bt cancels algebraically: grad_fx - proxy_grad = tanh(x@Wh+bh) @ Wt^T)
// ---------------------------------------------------------------------------
extern "C" void kernel_launch(void* const* d_in, const int* in_sizes, int n_in,
                              void* d_out, int out_size, void* d_ws, size_t ws_size,
                              hipStream_t stream) {
    const float* x  = (const float*)d_in[0];
    const float* Wh = (const float*)d_in[1];
    const float* bh = (const float*)d_in[2];
    const float* Wt = (const float*)d_in[3];
    // d_in[4] = bt : unused (cancels exactly in grad_fx - proxy_grad)

    char* ws = (char*)d_ws;
    float* Hp      = (float*)ws;                                        // 8*256*64*4 = 512 KB
    float* H       = (float*)(ws + (size_t)NSPLIT * B_DIM * K_DIM * 4); // 64 KB
    float* partial = H + B_DIM * K_DIM;                                 // 512*4 = 2 KB

    // Stage 1: H = tanh(X @ Wh + bh) via split-K fp32 WMMA
    senn_h_partial<<<dim3(B_DIM / 16, NSPLIT), 32, 0, stream>>>(x, Wh, Hp);
    senn_h_combine<<<(B_DIM * K_DIM + 255) / 256, 256, 0, stream>>>(Hp, bh, H);

    // Stage 2: per-strip sum of squares of H @ Wt^T
    senn_g_kernel<<<dim3(B_DIM / 16, D_DIM / 64), 32, 0, stream>>>(H, Wt, partial);

    // Stage 3: final deterministic reduction
    senn_reduce<<<1, 256, 0, stream>>>(partial, (float*)d_out,
                                       (B_DIM / 16) * (D_DIM / 64));
}